// RuleNet_57303453663343
// MI455X (gfx1250) — compile-verified
//
#include <hip/hip_runtime.h>
#include <math.h>

// RuleNet fuzzy-AND network for MI455X (gfx1250, wave32).
//   fit[b,c] = min_k fma(mu[c,k], xm1[b,k], 1.0)   with xm1 = concat(x0-1, -x0)
//   out[b]   = 5 + sum_c rw[c] * fit[b,c]          (V_WMMA_F32_16X16X4_F32 epilogue)
// Data movement: TDM (tensor_load_to_lds) stages the x0 tile, LDS holds the
// padded xm1 tile (stride 516 -> conflict-free ds_load_b128), mu rows stream
// through L2; row starts are prefetched once, outside all loops.

typedef float v2f __attribute__((ext_vector_type(2)));
typedef float v8f __attribute__((ext_vector_type(8)));
typedef unsigned int v4u __attribute__((ext_vector_type(4)));
typedef int v8i __attribute__((ext_vector_type(8)));
typedef int v4i __attribute__((ext_vector_type(4)));

#define RN_B 1024
#define RN_C 512
#define RN_V 256
#define RN_L 512          // 2V literals
#define XSTRIDE 516       // 516 % 64 == 4 -> conflict-free across 16 lanes

#if defined(__has_builtin)
#if __has_builtin(__builtin_amdgcn_tensor_load_to_lds) && __has_builtin(__builtin_amdgcn_s_wait_tensorcnt)
#define HAS_TDM 1
#else
#define HAS_TDM 0
#endif
#else
#define HAS_TDM 0
#endif

// Branchless sigmoid: native v_exp_f32 + v_rcp_f32 (no OCML range-reduction
// control flow polluting the inner loop).
__device__ __forceinline__ float sigmoid_fast(float c) {
    const float LOG2E = 1.44269504088896340736f;
    const float e = __builtin_amdgcn_exp2f(-c * LOG2E);
    return __builtin_amdgcn_rcpf(1.0f + e);
}

__global__ __launch_bounds__(256) void mu_precompute_kernel(const float4* __restrict__ conj,
                                                            float4* __restrict__ mu) {
    const int i = blockIdx.x * blockDim.x + threadIdx.x;   // 65536 float4s
    const float4 c = conj[i];
    float4 m;
    m.x = sigmoid_fast(c.x); m.y = sigmoid_fast(c.y);
    m.z = sigmoid_fast(c.z); m.w = sigmoid_fast(c.w);
    mu[i] = m;
}

// Block = 256 threads = 8 waves: 16 batches x (RN_C/NSPLIT) conjunctions.
// Wave w owns CPW conjunctions -> NCHUNK WMMA f32 16x16x4 accumulations.
// A-layout (32-bit 16x4): M = lane&15; lanes 0-15 hold K={0,1}, 16-31 K={2,3}.
// B = rw replicated over the 16 columns, so D rows carry the weighted sum.
template<bool PRE, int NSPLIT>
__global__ __launch_bounds__(256) void rulenet_kernel(const float* __restrict__ x0,
                                                      const float* __restrict__ muOrConj,
                                                      const float* __restrict__ rw,
                                                      float* __restrict__ outOrPart) {
    constexpr int CPB = RN_C / NSPLIT;   // conjunctions per block
    constexpr int CPW = CPB / 8;         // per wave
    constexpr int NCHUNK = CPW / 4;      // WMMAs per wave
    constexpr bool HOIST_B = (NCHUNK <= 4);
    constexpr int NPREF = (NCHUNK < 4) ? NCHUNK : 4;   // chunks to prefetch up front

    __shared__ float ldsX[16 * XSTRIDE];
    __shared__ float ldsRW[CPB];
    __shared__ float ldsP[8][16];
#if HAS_TDM
    __shared__ float ldsRaw[16 * RN_V];  // TDM destination: raw 16x256 x0 tile
#endif

    const int tid   = threadIdx.x;
    const int bbase = blockIdx.x * 16;
    const int cbase = blockIdx.y * CPB;

#if HAS_TDM
    if (tid < 32) {  // wave 0 issues the tensor DMA for the x0 tile
        const unsigned long long ga =
            (unsigned long long)(uintptr_t)(x0 + (size_t)bbase * RN_V);
        // Generic LDS pointer: low 32 bits are the LDS byte offset (aperture rule).
        const unsigned ldsOff = (unsigned)(uintptr_t)(void*)ldsRaw;
        v4u g0;
        g0.x = 1u;                                                 // count=1, user D#
        g0.y = ldsOff;                                             // lds_addr
        g0.z = (unsigned)ga;                                       // global_addr[31:0]
        g0.w = (unsigned)((ga >> 32) & 0x01FFFFFFu) | (2u << 30);  // addr[56:32], type=2
        v8i g1;
        g1[0] = (int)(2u << 16);                 // wg_mask=0, data_size=2 (4B)
        g1[1] = (int)((unsigned)RN_V << 16);     // tensor_dim0[15:0] = 256
        g1[2] = (int)((unsigned)RN_B << 16);     // tensor_dim0 hi=0, tensor_dim1 lo = 1024
        g1[3] = (int)((unsigned)RN_V << 16);     // tensor_dim1 hi=0, tile_dim0 = 256
        g1[4] = 16;                              // tile_dim1 = 16, tile_dim2 = 0
        g1[5] = RN_V;                            // tensor_dim0_stride = 256
        g1[6] = 0;                               // stride0 hi, stride1 lo
        g1[7] = 0;                               // stride1 hi
        const v4i g2 = {0, 0, 0, 0};
        const v4i g3 = {0, 0, 0, 0};
        const v8i gx = {0, 0, 0, 0, 0, 0, 0, 0}; // 6-arg toolchain: extra group, zero
        __builtin_amdgcn_tensor_load_to_lds(g0, g1, g2, g3, gx, 0);
        __builtin_amdgcn_s_wait_tensorcnt((short)0);
    }
    __syncthreads();  // tile visible to all waves
#endif

    // xm1 = (k < V) ? x0-1 : -x0, padded stride.  Uniform compile-time trip
    // count (scalar loop counter, no per-lane exit).
    #pragma unroll 1
    for (int i = 0; i < (16 * RN_L) / 256; ++i) {
        const int idx = tid + i * 256;
        const int m = idx >> 9;
        const int k = idx & (RN_L - 1);
#if HAS_TDM
        const float xv = ldsRaw[m * RN_V + (k & (RN_V - 1))];
#else
        const float xv = x0[(bbase + m) * RN_V + (k & (RN_V - 1))];
#endif
        ldsX[m * XSTRIDE + k] = (k < RN_V) ? (xv - 1.0f) : (0.0f - xv);
    }
    // rw staging: constexpr trip count; residual handled by one predicated store.
    #pragma unroll
    for (int i = 0; i < (CPB + 255) / 256; ++i) {
        const int c = tid + i * 256;
        if (c < CPB) ldsRW[c] = rw[cbase + c];
    }
    __syncthreads();

    const int wave = tid >> 5;
    const int lane = tid & 31;
    const int M    = lane & 15;   // A-matrix row (batch within tile)
    const int kg   = lane >> 4;   // 0 -> K={0,1}, 1 -> K={2,3}
    const int K0   = kg * 2;

    const int xoff  = M * XSTRIDE;
    const int cwave = cbase + wave * CPW;

    // Prefetch this wave's mu row starts once, OUTSIDE every loop (keeps
    // global_prefetch_b8 out of the hot loop's control-flow region).
    #pragma unroll
    for (int cc = 0; cc < NPREF; ++cc) {
        const float* __restrict__ p0 = muOrConj + (size_t)(cwave + cc * 4 + K0) * RN_L;
        __builtin_prefetch(p0, 0, 1);
        __builtin_prefetch(p0 + RN_L, 0, 1);
    }

    v8f acc = {};                 // WMMA C/D accumulator

    // One 4-conjunction chunk: min-reduce two mu rows against the x tile,
    // then fold fit into the WMMA accumulator with rw as the B operand.
    auto doChunk = [&](int c0, v2f Bm) {
        const float* __restrict__ r0 = muOrConj + (size_t)(c0 + K0) * RN_L;
        const float* __restrict__ r1 = r0 + RN_L;

        const float inf = __builtin_inff();
        float4 f0; f0.x = inf; f0.y = inf; f0.z = inf; f0.w = inf;
        float4 f1 = f0;

        #pragma unroll 4
        for (int k = 0; k < RN_L; k += 4) {
            const float4 xv = *reinterpret_cast<const float4*>(&ldsX[xoff + k]);
            float4 a0, a1;
            if (PRE) {
                a0 = *reinterpret_cast<const float4*>(r0 + k);
                a1 = *reinterpret_cast<const float4*>(r1 + k);
            } else {
                const float4 c0v = *reinterpret_cast<const float4*>(r0 + k);
                const float4 c1v = *reinterpret_cast<const float4*>(r1 + k);
                a0.x = sigmoid_fast(c0v.x); a0.y = sigmoid_fast(c0v.y);
                a0.z = sigmoid_fast(c0v.z); a0.w = sigmoid_fast(c0v.w);
                a1.x = sigmoid_fast(c1v.x); a1.y = sigmoid_fast(c1v.y);
                a1.z = sigmoid_fast(c1v.z); a1.w = sigmoid_fast(c1v.w);
            }
            f0.x = __builtin_fminf(f0.x, __builtin_fmaf(a0.x, xv.x, 1.0f));
            f0.y = __builtin_fminf(f0.y, __builtin_fmaf(a0.y, xv.y, 1.0f));
            f0.z = __builtin_fminf(f0.z, __builtin_fmaf(a0.z, xv.z, 1.0f));
            f0.w = __builtin_fminf(f0.w, __builtin_fmaf(a0.w, xv.w, 1.0f));
            f1.x = __builtin_fminf(f1.x, __builtin_fmaf(a1.x, xv.x, 1.0f));
            f1.y = __builtin_fminf(f1.y, __builtin_fmaf(a1.y, xv.y, 1.0f));
            f1.z = __builtin_fminf(f1.z, __builtin_fmaf(a1.z, xv.z, 1.0f));
            f1.w = __builtin_fminf(f1.w, __builtin_fmaf(a1.w, xv.w, 1.0f));
        }
        v2f Am;
        Am.x = __builtin_fminf(__builtin_fminf(f0.x, f0.y), __builtin_fminf(f0.z, f0.w));
        Am.y = __builtin_fminf(__builtin_fminf(f1.x, f1.y), __builtin_fminf(f1.z, f1.w));

        // D = A(16x4) x B(4x16) + C, EXEC all-ones here.
        acc = __builtin_amdgcn_wmma_f32_16x16x4_f32(
            false, Am, false, Bm, (short)0, acc, false, false);
    };

    if (HOIST_B) {
        // Pre-load all B operands (rw pairs) into registers; chunk loop fully
        // unrolled, so no ds_load / s_wait_dscnt sits in front of any WMMA.
        v2f Bmv[NCHUNK > 0 ? NCHUNK : 1];
        #pragma unroll
        for (int cc = 0; cc < NCHUNK; ++cc) {
            const int cl = wave * CPW + cc * 4;
            Bmv[cc].x = ldsRW[cl + K0];
            Bmv[cc].y = ldsRW[cl + K0 + 1];
        }
        #pragma unroll
        for (int cc = 0; cc < NCHUNK; ++cc) doChunk(cwave + cc * 4, Bmv[cc]);
    } else {
        #pragma unroll 1
        for (int cc = 0; cc < NCHUNK; ++cc) {
            const int cl = wave * CPW + cc * 4;
            v2f Bm; Bm.x = ldsRW[cl + K0]; Bm.y = ldsRW[cl + K0 + 1];
            doChunk(cwave + cc * 4, Bm);
        }
    }

    // Column 0 of D: lane 0 holds rows 0..7 (vgpr r -> M=r), lane 16 rows 8..15.
    if ((lane & 15) == 0) {
        const int rBase = kg * 8;
        #pragma unroll
        for (int r = 0; r < 8; ++r) ldsP[wave][rBase + r] = acc[r];
    }
    __syncthreads();

    // Deterministic cross-wave combine (fixed order, no FP atomics).
    if (tid < 16) {
        float s = 0.0f;
        #pragma unroll
        for (int w = 0; w < 8; ++w) s += ldsP[w][tid];
        if (NSPLIT == 1) outOrPart[bbase + tid] = 5.0f + s;
        else             outOrPart[(size_t)blockIdx.y * RN_B + bbase + tid] = s;
    }
}

// Deterministic cross-block combine for the NSPLIT=4 path.
__global__ __launch_bounds__(256) void combine_kernel(const float* __restrict__ part,
                                                      float* __restrict__ out) {
    const int i = blockIdx.x * blockDim.x + threadIdx.x;   // 1024 outputs
    float s = 5.0f;
    #pragma unroll
    for (int p = 0; p < 4; ++p) s += part[(size_t)p * RN_B + i];
    out[i] = s;
}

extern "C" void kernel_launch(void* const* d_in, const int* in_sizes, int n_in,
                              void* d_out, int out_size, void* d_ws, size_t ws_size,
                              hipStream_t stream) {
    const float* x0   = (const float*)d_in[0];   // (1024, 256)
    const float* conj = (const float*)d_in[1];   // (512, 512)
    const float* rw   = (const float*)d_in[2];   // (512,)
    float* out = (float*)d_out;                  // (1024,)

    const size_t muElems   = (size_t)RN_C * RN_L;
    const size_t muBytes   = muElems * sizeof(float);
    const size_t partBytes = (size_t)4 * RN_B * sizeof(float);

    if (ws_size >= muBytes + partBytes) {
        float* mu   = (float*)d_ws;
        float* part = (float*)d_ws + muElems;
        mu_precompute_kernel<<<(int)(muElems / 4 / 256), 256, 0, stream>>>(
            (const float4*)conj, (float4*)mu);
        rulenet_kernel<true, 4><<<dim3(RN_B / 16, 4), 256, 0, stream>>>(x0, mu, rw, part);
        combine_kernel<<<RN_B / 256, 256, 0, stream>>>(part, out);
    } else {
        // Workspace too small: single-pass fallback with inline branchless sigmoid.
        rulenet_kernel<false, 1><<<dim3(RN_B / 16, 1), 256, 0, stream>>>(x0, conj, rw, out);
    }
}